// SpikeFP32Sqrt_23407571764106
// MI455X (gfx1250) — compile-verified
//
#include <hip/hip_runtime.h>
#include <stdint.h>

typedef float    f4 __attribute__((ext_vector_type(4)));
typedef uint32_t u2 __attribute__((ext_vector_type(2)));

#define ROW_F   34              // padded LDS row stride in floats (136B, 8B-aligned, b64 conflict-free)
#define WAVE_F  (32 * ROW_F)    // floats of LDS per wave (32 rows)

// Bit-exact emulation of the reference's restoring-sqrt on an fp32 bit pattern.
__device__ __forceinline__ uint32_t bitsqrt_word(uint32_t u)
{
    const uint32_t s = u >> 31;
    const uint32_t e = (u >> 23) & 0xFFu;
    const uint32_t m = u & 0x7FFFFFu;

    // Normal path; reference treats e==0 as normal with implicit 1.
    const uint32_t S    = m | 0x800000u;
    const bool     Eodd = (e & 1u) == 0u;          // unbiased exponent odd (bias 127 odd)
    const uint32_t Seff = Eodd ? (S << 1) : S;     // <= 2^25 - 2
    const uint64_t X    = (uint64_t)Seff << 25;    // radicand, < 2^50

    float    rf = __builtin_amdgcn_sqrtf((float)Seff * 33554432.0f); // * 2^25
    uint32_t q  = (uint32_t)rf;
    if (q < (1u << 23))  q = 1u << 23;
    if (q > 0x1FFFFFEu)  q = 0x1FFFFFEu;
    while ((uint64_t)q * q > X) --q;
    while ((uint64_t)(q + 1) * (q + 1) <= X) ++q;  // q = floor(sqrt(X)), 25 bits

    const uint32_t sticky = ((uint64_t)q * q != X) ? 1u : 0u;
    const uint32_t rnd    = q & 1u;
    const uint32_t q24    = q >> 1;
    const uint32_t lsb    = q24 & 1u;
    const uint32_t mant   = ((q24 & 0x7FFFFFu) + (rnd & (sticky | lsb))) & 0x7FFFFFu;
    const uint32_t eo     = (e + (Eodd ? 126u : 127u)) >> 1;
    uint32_t res = (eo << 23) | mant;

    // Mux priority per reference:  inf > zero > (nan | negative) > normal
    const bool mzero = (m == 0u);
    if (s)                    res = 0x7FC00000u;   // negative non-zero -> qNaN
    if (e == 255u && !mzero)  res = 0x7FC00000u;   // NaN -> qNaN
    if (e == 0u   &&  mzero)  res = 0u;            // +-0 -> +0
    if (e == 255u &&  mzero)  res = 0x7F800000u;   // +-Inf -> +Inf (reference quirk)
    return res;
}

__global__ __launch_bounds__(256) void fp32_bitsqrt_kernel(
    const float* __restrict__ x, float* __restrict__ out, int nrows)
{
    __shared__ __align__(16) float lds[8 * WAVE_F];   // 8 waves * 32 rows * 34 floats = 34,816 B

    const uint32_t lane    = threadIdx.x & 31u;
    const uint32_t wid     = threadIdx.x >> 5;
    const size_t   gw      = (size_t)blockIdx.x * 8 + wid;   // global wave index
    const size_t   rowBase = gw * 32;                        // this wave's first row
    const size_t   myRow   = rowBase + lane;                 // this lane's row

    if (rowBase + 32 <= (size_t)nrows) {
        // ================= staged path (full 32-row wave tile) =================
        const uint32_t ldsWaveBase = (uint32_t)(uintptr_t)&lds[wid * WAVE_F];
        const uint32_t chunk       = (lane >> 4) * (ROW_F * 4u) + (lane & 15u) * 8u;
        const uint32_t tileByte    = (uint32_t)(rowBase * 128u);

        // ---- CDNA5 async staging: coalesced global -> LDS on ASYNCcnt ----
        // Async instruction i copies rows 2i..2i+1; lane chunks are globally
        // consecutive 8B (perfect coalescing) into the padded LDS layout.
        uint32_t lds_a = ldsWaveBase + chunk;
        uint32_t goff  = tileByte + lane * 8u;
#pragma unroll
        for (int i = 0; i < 16; ++i) {
            asm volatile("global_load_async_to_lds_b64 %0, %1, %2"
                         :: "v"(lds_a), "v"(goff), "s"(x)
                         : "memory");
            lds_a += 2u * ROW_F * 4u;
            goff  += 256u;
        }
        asm volatile("s_wait_asynccnt 0x0" ::: "memory");

        // ---- conflict-free b64 reads of this lane's row; pack 32 bit-floats ----
        u2* __restrict__ myrow = (u2*)&lds[wid * WAVE_F + lane * ROW_F];
        uint32_t u = 0u;
#pragma unroll
        for (int c = 0; c < 16; ++c) {
            u2 v = myrow[c];
            u |= ((v.x >> 23) & 1u) << (31 - (2 * c));      // 1.0f = 0x3F800000 -> bit23
            u |= ((v.y >> 23) & 1u) << (31 - (2 * c + 1));
        }

        const uint32_t res = bitsqrt_word(u);

        // ---- unpack result bits into the same LDS row (same-lane, same-wave:
        //      DS ops are in-order, so reuse is safe without a barrier) ----
#pragma unroll
        for (int c = 0; c < 16; ++c) {
            u2 w;
            w.x = ((res >> (31 - (2 * c)))     & 1u) ? 0x3F800000u : 0u;
            w.y = ((res >> (31 - (2 * c + 1))) & 1u) ? 0x3F800000u : 0u;
            myrow[c] = w;
        }
        // DS and async ops are not mutually ordered: drain the LDS writes
        // before the async store engine reads them.
        asm volatile("s_wait_dscnt 0x0" ::: "memory");

        // ---- CDNA5 async store: LDS -> coalesced global on ASYNCcnt ----
        // S_ENDPGM implies wait-idle, so no trailing s_wait_asynccnt is needed.
        uint32_t lds_s = ldsWaveBase + chunk;
        uint32_t soff  = tileByte + lane * 8u;
#pragma unroll
        for (int i = 0; i < 16; ++i) {
            asm volatile("global_store_async_from_lds_b64 %0, %1, %2"
                         :: "v"(soff), "v"(lds_s), "s"(out)
                         : "memory");
            lds_s += 2u * ROW_F * 4u;
            soff  += 256u;
        }
    } else {
        // ================= tail fallback (not hit for B = 1M rows) =================
        if (myRow >= (size_t)nrows) return;
        const f4* __restrict__ xr = reinterpret_cast<const f4*>(x) + myRow * 8;
        uint32_t u = 0u;
#pragma unroll
        for (int c = 0; c < 8; ++c) {
            f4 v = __builtin_nontemporal_load(xr + c);
#pragma unroll
            for (int k = 0; k < 4; ++k) {
                uint32_t b = (__float_as_uint(v[k]) >> 23) & 1u;
                u |= b << (31 - (c * 4 + k));
            }
        }
        const uint32_t res = bitsqrt_word(u);
        f4* __restrict__ orw = reinterpret_cast<f4*>(out) + myRow * 8;
#pragma unroll
        for (int c = 0; c < 8; ++c) {
            f4 o;
#pragma unroll
            for (int k = 0; k < 4; ++k)
                o[k] = (float)((res >> (31 - (c * 4 + k))) & 1u);
            __builtin_nontemporal_store(o, orw + c);
        }
    }
}

extern "C" void kernel_launch(void* const* d_in, const int* in_sizes, int n_in,
                              void* d_out, int out_size, void* d_ws, size_t ws_size,
                              hipStream_t stream)
{
    const float* x   = (const float*)d_in[0];
    float*       out = (float*)d_out;
    const int nrows  = in_sizes[0] / 32;             // 32 bit-floats per row
    const int nwaves = (nrows + 31) / 32;            // one 32-row tile per wave
    const int blocks = (nwaves + 7) / 8;             // 8 waves (256 threads) per block
    fp32_bitsqrt_kernel<<<blocks, 256, 0, stream>>>(x, out, nrows);
}